// VoxelEncoder_39444979646669
// MI455X (gfx1250) — compile-verified
//
#include <hip/hip_runtime.h>

#define GXD 32
#define GYD 24
#define GTD 5
#define N_BINS (2 * GTD * GYD * GXD)   // 7680
#define BLK 256

typedef float v2f __attribute__((ext_vector_type(2)));
typedef float v8f __attribute__((ext_vector_type(8)));

// ---------------------------------------------------------------- init
__global__ void voxel_init_kernel(float* __restrict__ out, unsigned* __restrict__ ws) {
    int i = blockIdx.x * blockDim.x + threadIdx.x;
    if (i < N_BINS) out[i] = 0.0f;
    if (i == 0) {
        ws[0] = 0x7F800000u;  // +inf bits (t >= 0, uint order == float order)
        ws[1] = 0x00000000u;  // 0.0f bits
    }
}

// ---------------------------------------------------------------- pass 1: min/max of t
__global__ void __launch_bounds__(BLK)
voxel_minmax_kernel(const float4* __restrict__ ev, int n4, unsigned* __restrict__ ws) {
    int tid    = blockIdx.x * blockDim.x + threadIdx.x;
    int stride = gridDim.x * blockDim.x;
    float tmn = __int_as_float(0x7F800000);   // +inf
    float tmx = 0.0f;                         // t >= 0
    for (int i = tid; i < n4; i += stride) {
        float4 e = ev[i];                     // global_load_b128
        tmn = fminf(tmn, e.z);
        tmx = fmaxf(tmx, e.z);
    }
    // wave32 butterfly reduction
    #pragma unroll
    for (int m = 16; m >= 1; m >>= 1) {
        tmn = fminf(tmn, __shfl_xor(tmn, m, 32));
        tmx = fmaxf(tmx, __shfl_xor(tmx, m, 32));
    }
    if ((threadIdx.x & 31) == 0) {
        atomicMin(&ws[0], __float_as_uint(tmn));
        atomicMax(&ws[1], __float_as_uint(tmx));
    }
}

// ---------------------------------------------------------------- pass 2: LDS-privatized histogram
__global__ void __launch_bounds__(BLK)
voxel_hist_kernel(const float4* __restrict__ ev, int n4,
                  const unsigned* __restrict__ ws, float* __restrict__ out) {
    __shared__ unsigned h[N_BINS];            // 30 KB of 320 KB/WGP
    for (int b = threadIdx.x; b < N_BINS; b += BLK) h[b] = 0u;
    __syncthreads();

    const float tmn   = __uint_as_float(ws[0]);
    const float tmx   = __uint_as_float(ws[1]);
    const float range = tmx - tmn;
    const bool  renorm = (tmx > tmn);
    const float xs = (float)GXD / 640.0f;     // matches jnp f32 constant
    const float ys = (float)GYD / 480.0f;
    const float ts = (float)GTD / 50.0f;

    int tid    = blockIdx.x * blockDim.x + threadIdx.x;
    int stride = gridDim.x * blockDim.x;
    for (int i = tid; i < n4; i += stride) {
        float4 e = ev[i];                     // x, y, t, p
        int xv = min(max((int)(e.x * xs), 0), GXD - 1);
        int yv = min(max((int)(e.y * ys), 0), GYD - 1);
        float tt = renorm ? ((e.z - tmn) / range) * 50.0f : e.z;  // IEEE f32 div, as jnp
        int tv = min(max((int)(tt * ts), 0), GTD - 1);
        int pidx = (e.w > 0.0f) ? 0 : 1;
        int flat = ((pidx * GTD + tv) * GYD + yv) * GXD + xv;
        atomicAdd(&h[flat], 1u);              // ds_add_u32
    }
    __syncthreads();

    for (int b = threadIdx.x; b < N_BINS; b += BLK) {
        unsigned c = h[b];
        if (c) atomicAdd(&out[b], (float)c);  // global_atomic_add_f32 (exact: counts < 2^24)
    }
}

// ---------------------------------------------------------------- normalize (WMMA cross-lane sum)
__global__ void __launch_bounds__(BLK)
voxel_norm_kernel(float* __restrict__ out) {
    __shared__ float part[BLK];
    __shared__ float s_total;
    const int tid = threadIdx.x;

    float v[N_BINS / BLK];                    // 30 bins per thread
    float local = 0.0f;
    #pragma unroll
    for (int j = 0; j < N_BINS / BLK; ++j) {
        v[j] = out[tid + j * BLK];
        local += v[j];
    }
    part[tid] = local;
    __syncthreads();

    if (tid < 32) {                           // wave 0, full EXEC for WMMA
        float p = 0.0f;
        #pragma unroll
        for (int w = 0; w < 8; ++w) p += part[tid + w * 32];
        // 32-lane sum via V_WMMA_F32_16X16X4_F32:
        // A(16x4): lane L<16 holds {A[L][0],A[L][1]} = {p_L, 0};
        //          lane L+16 holds {A[L][2],A[L][3]} = {p_{L+16}, 0}.
        // B(4x16) = all ones  =>  D[m][n] = p_m + p_{m+16}.
        v2f a = { p, 0.0f };
        v2f b = { 1.0f, 1.0f };
        v8f c = {};
        c = __builtin_amdgcn_wmma_f32_16x16x4_f32(
                /*neg_a=*/false, a, /*neg_b=*/false, b,
                /*c_mod=*/(short)0, c, /*reuse_a=*/false, /*reuse_b=*/false);
        // Lane n sums D rows 0..7, lane n+16 sums rows 8..15; xor-16 completes.
        float s = c[0] + c[1] + c[2] + c[3] + c[4] + c[5] + c[6] + c[7];
        s += __shfl_xor(s, 16, 32);
        if (tid == 0) s_total = s;
    }
    __syncthreads();

    const float s = s_total;                  // s > 0 always (N events counted)
    #pragma unroll
    for (int j = 0; j < N_BINS / BLK; ++j)
        out[tid + j * BLK] = v[j] / s;
}

// ---------------------------------------------------------------- launcher
extern "C" void kernel_launch(void* const* d_in, const int* in_sizes, int n_in,
                              void* d_out, int out_size, void* d_ws, size_t ws_size,
                              hipStream_t stream) {
    const float4* ev = (const float4*)d_in[0];   // [N,4]: x,y,t,p per row
    const int n4 = in_sizes[0] / 4;              // number of events
    float* out = (float*)d_out;                  // 7680 floats
    unsigned* ws = (unsigned*)d_ws;              // [0]=min bits, [1]=max bits

    voxel_init_kernel<<<(N_BINS + BLK - 1) / BLK, BLK, 0, stream>>>(out, ws);
    voxel_minmax_kernel<<<2048, BLK, 0, stream>>>(ev, n4, ws);
    voxel_hist_kernel<<<2048, BLK, 0, stream>>>(ev, n4, ws, out);
    voxel_norm_kernel<<<1, BLK, 0, stream>>>(out);
}